// AnchorLoss_47132971106501
// MI455X (gfx1250) — compile-verified
//
#include <hip/hip_runtime.h>
#include <hip/hip_bf16.h>

// AnchorLoss on MI455X (gfx1250), compile-only tuned.
//
// loss = -(2*n*sum(a^2) - 2*||sum_rows(a)||^2) / sqrt(768) / (k*k)
// a: [8192, 6144] f32 (anchors reshaped), 201 MB -> one streaming pass,
// bandwidth-bound (~8.6 us at 23.3 TB/s). Column sums are produced with
// V_WMMA_F32_16X16X4_F32 (A = ones 16x4, B = 4x16 data tile): D accumulates
// per-column sums, replicated over M, so every lane's acc[0] holds the
// partial column sum for its column. sum(a^2) rides along in VALU FMAs.

typedef float v2f __attribute__((ext_vector_type(2)));
typedef float v8f __attribute__((ext_vector_type(8)));

#define NROWS 8192
#define NCOLS 6144          // 8 * 768
#define RCHUNKS 16          // row chunks (parallelism across rows)
#define ROWS_PER_CHUNK (NROWS / RCHUNKS)   // 512
#define COLGROUPS (NCOLS / 16)             // 384 groups of 16 columns
#define WAVES_PER_BLOCK 8
#define BLOCK_THREADS 256
#define NBLOCKS1 (COLGROUPS * RCHUNKS / WAVES_PER_BLOCK)  // 768

__global__ __launch_bounds__(BLOCK_THREADS)
void anchor_stage1(const float* __restrict__ A,
                   float* __restrict__ colPart,   // [RCHUNKS][NCOLS]
                   float* __restrict__ ssPart)    // [NBLOCKS1]
{
    const int tid  = threadIdx.x;
    const int lane = tid & 31;
    const int wave = tid >> 5;

    // wave id -> (row chunk, column group); adjacent waves in a block share
    // the same rows and touch adjacent 16-col groups (good L2 locality).
    const int gid   = blockIdx.x * WAVES_PER_BLOCK + wave;  // 0..6143
    const int chunk = gid / COLGROUPS;                      // 0..15
    const int cg    = gid % COLGROUPS;                      // 0..383
    const int col0  = cg * 16;
    const int n     = lane & 15;            // column within group (B: N=lane)
    const int kb    = (lane >> 4) << 1;     // B fragment: lanes 0-15 hold K=0,1; 16-31 hold K=2,3
    const int row0  = chunk * ROWS_PER_CHUNK;

    v2f ones; ones.x = 1.0f; ones.y = 1.0f; // A = 16x4 all-ones fragment
    v8f acc = {};                            // C/D accumulator (column sums)
    float ss = 0.0f;                         // per-lane sum of squares

    const float* p = A + (size_t)(row0 + kb) * NCOLS + col0 + n;

    #pragma unroll 4
    for (int r = 0; r < ROWS_PER_CHUNK; r += 4) {
        float v0 = p[0];          // row r+kb,   column col0+n (64B-aligned coalesced)
        float v1 = p[NCOLS];      // row r+kb+1, column col0+n
        p += 4 * NCOLS;
        ss = fmaf(v0, v0, ss);
        ss = fmaf(v1, v1, ss);
        v2f b; b.x = v0; b.y = v1;
        // D = ones(16x4) x B(4x16) + D  ->  column sums accumulate
        acc = __builtin_amdgcn_wmma_f32_16x16x4_f32(
            /*neg_a=*/false, ones, /*neg_b=*/false, b,
            /*c_mod=*/(short)0, acc, /*reuse_a=*/false, /*reuse_b=*/false);
    }

    // Every M-row of D is identical (all-ones A); lane L's acc[0] is the
    // column-sum partial for column col0+(L&15). Store from lanes 0-15.
    if (lane < 16)
        colPart[chunk * NCOLS + col0 + n] = acc[0];

    // Block-level reduction of sum-of-squares.
    __shared__ float red[BLOCK_THREADS];
    red[tid] = ss;
    __syncthreads();
    for (int s = BLOCK_THREADS / 2; s > 0; s >>= 1) {
        if (tid < s) red[tid] += red[tid + s];
        __syncthreads();
    }
    if (tid == 0) ssPart[blockIdx.x] = red[0];
}

__global__ __launch_bounds__(BLOCK_THREADS)
void anchor_stage2(const float* __restrict__ colPart,
                   const float* __restrict__ ssPart,
                   float* __restrict__ out)
{
    const int tid = threadIdx.x;

    // dot = sum over columns of (sum over chunks of partial)^2
    float dot = 0.0f;
    for (int c = tid; c < NCOLS; c += BLOCK_THREADS) {
        float s = 0.0f;
        #pragma unroll
        for (int ch = 0; ch < RCHUNKS; ++ch)
            s += colPart[ch * NCOLS + c];
        dot = fmaf(s, s, dot);
    }

    float ss = 0.0f;
    for (int b = tid; b < NBLOCKS1; b += BLOCK_THREADS)
        ss += ssPart[b];

    __shared__ float rd[BLOCK_THREADS];
    __shared__ float rs[BLOCK_THREADS];
    rd[tid] = dot;
    rs[tid] = ss;
    __syncthreads();
    for (int s = BLOCK_THREADS / 2; s > 0; s >>= 1) {
        if (tid < s) { rd[tid] += rd[tid + s]; rs[tid] += rs[tid + s]; }
        __syncthreads();
    }
    if (tid == 0) {
        float pair = 2.0f * (float)NROWS * rs[0] - 2.0f * rd[0];
        const float factor = 27.712812921102035f;  // sqrt(768)
        out[0] = -(pair / factor) / 64.0f;          // / (k*k), k=8
    }
}

extern "C" void kernel_launch(void* const* d_in, const int* in_sizes, int n_in,
                              void* d_out, int out_size, void* d_ws, size_t ws_size,
                              hipStream_t stream) {
    const float* A   = (const float*)d_in[0];   // anchors, f32, 8192*8*768
    float*       out = (float*)d_out;           // scalar f32

    // Workspace layout: [RCHUNKS*NCOLS] column-sum partials, then [NBLOCKS1]
    // sum-of-squares partials. Total ~396 KB.
    float* colPart = (float*)d_ws;
    float* ssPart  = colPart + (size_t)RCHUNKS * NCOLS;

    anchor_stage1<<<NBLOCKS1, BLOCK_THREADS, 0, stream>>>(A, colPart, ssPart);
    anchor_stage2<<<1, BLOCK_THREADS, 0, stream>>>(colPart, ssPart, out);
}